// SAGEConv_57492432224406
// MI455X (gfx1250) — compile-verified
//
#include <hip/hip_runtime.h>

typedef __attribute__((ext_vector_type(2))) float v2f;
typedef __attribute__((ext_vector_type(4))) float v4f;
typedef __attribute__((ext_vector_type(8))) float v8f;
typedef __attribute__((ext_vector_type(4))) unsigned int v4u;
typedef __attribute__((ext_vector_type(8))) int v8i;
typedef __attribute__((ext_vector_type(4))) int v4i;

constexpr int F_IN  = 128;
constexpr int F_OUT = 128;
constexpr int KDIM  = 256;      // 2*F_IN
constexpr int ROWS  = 64;       // target rows per block
constexpr int LDSW  = F_IN + 4; // 132; 132 % 64 == 4 -> A-frag ds_load_b64 conflict-free

__global__ __launch_bounds__(256, 2)
void sage_fused_wmma(const int* __restrict__ rowptr,
                     const int* __restrict__ colind,
                     const float* __restrict__ xnbr,
                     const float* __restrict__ xtgt,
                     const float* __restrict__ W,       // [F_OUT][KDIM] row-major
                     const float* __restrict__ lin_bias,
                     const float* __restrict__ bias2,
                     float* __restrict__ out,           // [n_tgt][F_OUT]
                     int n_tgt)
{
    __shared__ float catT[ROWS * LDSW]; // x_target half of cat (K = 0..127), filled by TDM
    __shared__ float catA[ROWS * LDSW]; // mean-agg half of cat (K = 128..255)

    const int tid  = threadIdx.x;
    const int lane = tid & 31;
    const int wave = tid >> 5;
    const int row0 = blockIdx.x * ROWS;

    // ------------- Phase 0: TDM async load of the x_target tile into catT -------------
    // 2D descriptor: 128-wide f32 rows, 64-row tile, tensor_dim1 = remaining rows so the
    // partial last block reads zeros (TDM OOB-read semantics). LDS padding of 4 dwords
    // after every 128 dwords lands each row at stride LDSW=132.
    if (wave == 0) {
        const unsigned long long ga = (unsigned long long)(const void*)(xtgt + (size_t)row0 * F_IN);
        const unsigned int ldsaddr  = (unsigned int)(unsigned long long)(const void*)catT;
        const unsigned int rrem     = (unsigned int)(n_tgt - row0);

        v4u g0;
        g0[0] = 1u;                                       // count=1, user descriptor
        g0[1] = ldsaddr;                                  // lds_addr
        g0[2] = (unsigned int)ga;                         // global_addr[31:0]
        g0[3] = (unsigned int)((ga >> 32) & 0x01FFFFFFull)
              | (2u << 30);                               // global_addr[56:32] | type=2 (image)

        v8i g1;
        g1[0] = (int)((2u << 16)    // data_size = 4B
                    | (1u << 20)    // pad_enable
                    | (6u << 22)    // pad_interval: 128 dwords
                    | (3u << 25));  // pad_amount: 4 dwords
        g1[1] = (int)(128u << 16);                        // tensor_dim0[15:0] = 128
        g1[2] = (int)((rrem & 0xffffu) << 16);            // tensor_dim1[15:0]
        g1[3] = (int)((rrem >> 16) | (128u << 16));       // tensor_dim1[31:16] | tile_dim0=128
        g1[4] = 64;                                       // tile_dim1=64, tile_dim2=0
        g1[5] = 128;                                      // tensor_dim0_stride = 128
        g1[6] = 0;
        g1[7] = 0;

        const v4i gz4 = {0, 0, 0, 0};
        const v8i gz8 = {0, 0, 0, 0, 0, 0, 0, 0};
        __builtin_amdgcn_tensor_load_to_lds(g0, g1, gz4, gz4, gz8, 0);
    }

    // ------------- Phase 1: mean aggregation into catA (overlaps the TDM) -------------
    // wave w handles local rows [w*8, w*8+8)
    for (int i = 0; i < 8; ++i) {
        const int lr = wave * 8 + i;
        const int gr = row0 + lr;
        float ax = 0.f, ay = 0.f, az = 0.f, aw = 0.f;
        if (gr < n_tgt) {
            const int ebase = rowptr[gr];
            const int cnt   = rowptr[gr + 1] - ebase;
            int myidx = (lane < cnt) ? colind[ebase + lane] : 0;
            const int c32 = cnt < 32 ? cnt : 32;
            for (int j = 0; j < c32; ++j) {
                const int col = __shfl(myidx, j, 32);
                const v4f v = *(const v4f*)(xnbr + (size_t)col * F_IN + lane * 4);
                ax += v.x; ay += v.y; az += v.z; aw += v.w;
            }
            for (int j = 32; j < cnt; ++j) { // generality; never taken for DEG=16
                const int col = colind[ebase + j];
                const v4f v = *(const v4f*)(xnbr + (size_t)col * F_IN + lane * 4);
                ax += v.x; ay += v.y; az += v.z; aw += v.w;
            }
            const float inv = 1.0f / (float)(cnt > 0 ? cnt : 1);
            ax *= inv; ay *= inv; az *= inv; aw *= inv;
        }
        v4f av; av.x = ax; av.y = ay; av.z = az; av.w = aw;
        *(v4f*)(&catA[lr * LDSW] + lane * 4) = av;
    }

    if (wave == 0) __builtin_amdgcn_s_wait_tensorcnt(0); // TDM done before barrier orders it for all
    __syncthreads();

    // ------------- Phase 2: [64 x 256] @ [256 x 128] via V_WMMA_F32_16X16X4_F32 -------
    // wave w owns output cols [w*16, w*16+16); 4 row sub-tiles of 16 share each B-frag.
    const int obase = wave * 16;
    const int m     = lane & 15;        // A row / B col within fragment
    const int khalf = (lane >> 4) << 1; // upper half-wave carries K+2,K+3
    const float bsum = lin_bias[obase + m] + bias2[obase + m];

    v8f acc[4] = {};

    // B[k][n] = W[obase+n][k]; lane layout mirrors A (VGPR0: K/K+2, VGPR1: K+1/K+3)
    const float* wrow = W + (size_t)(obase + m) * KDIM + khalf;

    #pragma unroll 4
    for (int k0 = 0; k0 < F_IN; k0 += 4) {           // K = 0..127 from catT
        const v2f b = *(const v2f*)(wrow + k0);
        #pragma unroll
        for (int rt = 0; rt < 4; ++rt) {
            const v2f a = *(const v2f*)&catT[(rt * 16 + m) * LDSW + k0 + khalf];
            acc[rt] = __builtin_amdgcn_wmma_f32_16x16x4_f32(
                false, a, false, b, (short)0, acc[rt], false, false);
        }
    }
    #pragma unroll 4
    for (int k0 = 0; k0 < F_IN; k0 += 4) {           // K = 128..255 from catA
        const v2f b = *(const v2f*)(wrow + F_IN + k0);
        #pragma unroll
        for (int rt = 0; rt < 4; ++rt) {
            const v2f a = *(const v2f*)&catA[(rt * 16 + m) * LDSW + k0 + khalf];
            acc[rt] = __builtin_amdgcn_wmma_f32_16x16x4_f32(
                false, a, false, b, (short)0, acc[rt], false, false);
        }
    }

    // ------------- Epilogue: C/D layout -> global NT stores -------------
    // VGPR v of acc[rt] holds row (rt*16 + v + 8*(lane>=16)), col (obase + (lane&15)).
    const int mhi = (lane >> 4) << 3;
    float* pout = out + (size_t)(row0 + mhi) * F_OUT + obase + m;
    if (row0 + ROWS <= n_tgt) {
        // full tile (1562 of 1563 blocks): unguarded stores, immediate offsets
        #pragma unroll
        for (int rt = 0; rt < 4; ++rt)
            #pragma unroll
            for (int v = 0; v < 8; ++v)
                __builtin_nontemporal_store(acc[rt][v] + bsum,
                                            pout + (size_t)((rt * 16 + v) * F_OUT));
    } else {
        #pragma unroll
        for (int rt = 0; rt < 4; ++rt)
            #pragma unroll
            for (int v = 0; v < 8; ++v)
                if (row0 + rt * 16 + v + mhi < n_tgt)
                    __builtin_nontemporal_store(acc[rt][v] + bsum,
                                                pout + (size_t)((rt * 16 + v) * F_OUT));
    }
}

extern "C" void kernel_launch(void* const* d_in, const int* in_sizes, int n_in,
                              void* d_out, int out_size, void* d_ws, size_t ws_size,
                              hipStream_t stream) {
    const int*   rowptr = (const int*)d_in[0];   // csr_row_ptr  [n_tgt+1]
    const int*   colind = (const int*)d_in[1];   // csr_col_ind  [n_edge]
    // d_in[2] = sample_count (unused by the reference math)
    const float* xnbr   = (const float*)d_in[3]; // x_neighboor  [n_nbr][128]
    const float* xtgt   = (const float*)d_in[4]; // x_target     [n_tgt][128]
    const float* W      = (const float*)d_in[5]; // lin_weight   [128][256]
    const float* lb     = (const float*)d_in[6]; // lin_bias     [128]
    const float* bb     = (const float*)d_in[7]; // bias         [128]
    float*       out    = (float*)d_out;

    const int n_tgt  = in_sizes[0] - 1;
    const int blocks = (n_tgt + ROWS - 1) / ROWS;
    sage_fused_wmma<<<blocks, 256, 0, stream>>>(rowptr, colind, xnbr, xtgt, W, lb, bb, out, n_tgt);
}